// Transform_86466281603245
// MI455X (gfx1250) — compile-verified
//
#include <hip/hip_runtime.h>

// CDNA5 / gfx1250: wave32, WMMA f32 16x16x4.
typedef __attribute__((ext_vector_type(2))) float v2f;
typedef __attribute__((ext_vector_type(8))) float v8f;

#define SCL (62.0f / 63.0f)   // 0.5*(t+1)*(dim-2) with t = -1 + 2i/(dim-1)  ->  i * (dim-2)/(dim-1)

// One wave32 handles one strip: fixed (b, d, h), 16 consecutive w voxels, all 32 channels.
// out[16x32] = A[16x68] x V[68x32] done as 17 chained V_WMMA_F32_16X16X4_F32 per 16-ch tile.
__global__ __launch_bounds__(256)
void trilerp_wmma_kernel(const float* __restrict__ img, float* __restrict__ out) {
    const int lane = threadIdx.x & 31;
    const int wid  = (blockIdx.x << 3) + (threadIdx.x >> 5);   // 65536 waves total

    const int ws = wid & 3;            // w-strip index (16 voxels each)
    const int h  = (wid >> 2) & 63;
    const int d  = (wid >> 8) & 63;
    const int b  = wid >> 14;

    const int w0 = ws << 4;
    const int m  = lane & 15;          // voxel row within strip / channel col within tile
    const int hi = lane >> 4;          // K-half select (A holds K=2*hi, 2*hi+1)

    // ---- wave-uniform y/z interpolation ----
    const float zsf = (float)d * SCL;
    const int   z0  = (int)zsf;                  // zsf >= 0
    const float fz  = zsf - (float)z0;
    int z1 = z0 + 1; if (z1 > 63) z1 = 63;

    const float ysf = (float)h * SCL;
    const int   y0  = (int)ysf;
    const float fy  = ysf - (float)y0;
    int y1 = y0 + 1; if (y1 > 63) y1 = 63;

    // ---- per-voxel (per-m) x interpolation ----
    const float xsf = (float)(w0 + m) * SCL;
    const int   x0m = (int)xsf;
    const float fx  = xsf - (float)x0m;

    const int x_base = (int)((float)w0 * SCL);   // uniform; x-columns span [x_base, x_base+16]

    // yz corner weights for this lane's two K rows:
    //   q0=(z0,y0) q1=(z0,y1) q2=(z1,y0) q3=(z1,y1); lane covers q=2*hi, 2*hi+1
    const float wz  = hi ? fz : (1.0f - fz);
    const float wqa = wz * (1.0f - fy);
    const float wqb = wz * fy;

    // B-matrix feed pointers: rows (zsel,y0) and (zsel,y1), column channel = m (+16 for tile1)
    const int zsel = hi ? z1 : z0;
    const float* p0 = img + ((((long)b * 64 + zsel) * 64 + y0) * 64 + x_base) * 32 + m;
    const float* p1 = img + ((((long)b * 64 + zsel) * 64 + y1) * 64 + x_base) * 32 + m;

    v8f acc0 = {};   // channels 0..15
    v8f acc1 = {};   // channels 16..31

#pragma unroll
    for (int xi = 0; xi < 17; ++xi) {
        // x-weight for this lane's voxel at column x_base+xi
        const int   dxi = (x_base + xi) - x0m;
        const float wx  = (dxi == 0) ? (1.0f - fx) : ((dxi == 1) ? fx : 0.0f);

        v2f a;                        // A 16x4 slab: lanes0-15 K={0,1}, lanes16-31 K={2,3}
        a.x = wx * wqa;
        a.y = wx * wqb;

        v2f b0, b1;                   // B 4x16 slab: lane half selects K rows {2hi, 2hi+1}
        b0.x = p0[0];   b0.y = p1[0];
        b1.x = p0[16];  b1.y = p1[16];

        acc0 = __builtin_amdgcn_wmma_f32_16x16x4_f32(false, a, false, b0, (short)0, acc0, false, false);
        acc1 = __builtin_amdgcn_wmma_f32_16x16x4_f32(false, a, false, b1, (short)0, acc1, false, false);

        p0 += 32;                     // next x column (32 floats)
        p1 += 32;
    }

    // ---- store D (16x16 f32 layout: VGPR r -> M = r + 8*hi, N = m) ----
    // Non-temporal: output is write-once; keep the 128MB input resident in L2.
    float* o = out + ((((long)b * 64 + d) * 64 + h) * 64 + w0) * 32 + (long)hi * 8 * 32 + m;
#pragma unroll
    for (int r = 0; r < 8; ++r) {
        __builtin_nontemporal_store(acc0[r], o + r * 32);
        __builtin_nontemporal_store(acc1[r], o + r * 32 + 16);
    }
}

extern "C" void kernel_launch(void* const* d_in, const int* in_sizes, int n_in,
                              void* d_out, int out_size, void* d_ws, size_t ws_size,
                              hipStream_t stream) {
    (void)in_sizes; (void)n_in; (void)d_ws; (void)ws_size; (void)out_size;
    const float* x = (const float*)d_in[0];   // (4,64,64,64,32) f32
    float* out = (float*)d_out;               // (4,64,64,64,32) f32

    // 65536 wave32 strips = 8192 blocks x 256 threads (exact cover -> EXEC all-ones for WMMA)
    trilerp_wmma_kernel<<<dim3(8192), dim3(256), 0, stream>>>(x, out);
}